// GNN_73418170958393
// MI455X (gfx1250) — compile-verified
//
#include <hip/hip_runtime.h>
#include <hip/hip_bf16.h>
#include <math.h>

#define NN   50000
#define EE   400000
#define GG   64
#define IND  64
#define HIDD 128
#define NH   4
#define E2   (EE + NN)

typedef float v2f __attribute__((ext_vector_type(2)));
typedef float v8f __attribute__((ext_vector_type(8)));

__device__ __forceinline__ float leaky(float x, float s) { return x > 0.f ? x : s * x; }

// monotonic float <-> uint map for atomicMax on floats
__device__ __forceinline__ unsigned f2o(float x) {
  unsigned b = __float_as_uint(x);
  return (b & 0x80000000u) ? ~b : (b | 0x80000000u);
}
__device__ __forceinline__ float o2f(unsigned o) {
  unsigned b = (o & 0x80000000u) ? (o & 0x7fffffffu) : ~o;
  return __uint_as_float(b);
}

// ---------------- generic fills ----------------
__global__ void k_fill_f32(float* __restrict__ p, float v, int n) {
  int i = blockIdx.x * blockDim.x + threadIdx.x;
  if (i < n) p[i] = v;
}
__global__ void k_fill_u32(unsigned* __restrict__ p, unsigned v, int n) {
  int i = blockIdx.x * blockDim.x + threadIdx.x;
  if (i < n) p[i] = v;
}
// broadcast 128-wide bias over rows
__global__ void k_fill_bias(float* __restrict__ p, const float* __restrict__ b, int n) {
  int i = blockIdx.x * blockDim.x + threadIdx.x;
  if (i < n) p[i] = b[i & (HIDD - 1)];
}

// ---------------- graph / degree prep ----------------
__global__ void k_count_dst(const int* __restrict__ ei, unsigned* __restrict__ cnt) {
  int i = blockIdx.x * blockDim.x + threadIdx.x;
  if (i < EE) atomicAdd(&cnt[ei[EE + i]], 1u);
}
__global__ void k_count_graph(const int* __restrict__ batch, unsigned* __restrict__ gc) {
  int i = blockIdx.x * blockDim.x + threadIdx.x;
  if (i < NN) atomicAdd(&gc[batch[i]], 1u);
}
__global__ void k_scan_graph(const unsigned* __restrict__ gc, int* __restrict__ goff) {
  if (threadIdx.x == 0 && blockIdx.x == 0) {
    int a = 0;
    for (int g = 0; g < GG; ++g) { goff[g] = a; a += (int)gc[g]; }
    goff[GG] = a;
  }
}

// ---------------- self_attr = segment_mean(edge_attr, dst) ----------------
__global__ void k_self_accum(const float* __restrict__ ea, const int* __restrict__ ei,
                             float* __restrict__ sum) {
  int i = blockIdx.x * blockDim.x + threadIdx.x;
  if (i < EE * HIDD) {
    int e = i >> 7, c = i & (HIDD - 1);
    atomicAdd(&sum[ei[EE + e] * HIDD + c], ea[i]);
  }
}
__global__ void k_self_fin(float* __restrict__ sum, const unsigned* __restrict__ cnt,
                           float* __restrict__ dinv) {
  int i = blockIdx.x * blockDim.x + threadIdx.x;
  if (i < NN * HIDD) {
    int n = i >> 7;
    float c = fmaxf((float)cnt[n], 1.f);
    sum[i] = sum[i] / c;
    if ((i & (HIDD - 1)) == 0) dinv[n] = rsqrtf((float)(cnt[n] + 1u)); // deg includes self loop
  }
}

// ---------------- v_e[k][h] = sum_c We[k][h*128+c] * att_edge[h][c] ----------------
__global__ void k_ve(const float* __restrict__ We, const float* __restrict__ atte,
                     float* __restrict__ ve) {
  int i = blockIdx.x * blockDim.x + threadIdx.x;
  if (i < HIDD * NH) {
    int k = i >> 2, h = i & 3;
    float s = 0.f;
    for (int c = 0; c < HIDD; ++c) s += We[k * (NH * HIDD) + h * HIDD + c] * atte[h * HIDD + c];
    ve[i] = s;
  }
}

// ---------------- fp32 WMMA GEMM: C[MxN] = A[MxK] * B[KxN], row-major ----------------
// M%16==0, N%16==0, K%4==0. One wave -> one 16x16 tile, V_WMMA_F32_16X16X4_F32 along K.
__global__ void k_wmma_gemm_f32(const float* __restrict__ A, const float* __restrict__ B,
                                float* __restrict__ C, int M, int Nn, int K) {
  int wave = (blockIdx.x * blockDim.x + threadIdx.x) >> 5;
  int lane = threadIdx.x & 31;
  int ntiles = Nn >> 4;
  int total = (M >> 4) * ntiles;
  if (wave >= total) return;
  int m0 = (wave / ntiles) << 4;
  int n0 = (wave % ntiles) << 4;
  int row = lane & 15;           // M index (A) / N index (B,C)
  int kh  = (lane >> 4) << 1;    // K sub-offset per half-wave
  v8f acc = {0.f, 0.f, 0.f, 0.f, 0.f, 0.f, 0.f, 0.f};
  const float* arow = A + (m0 + row) * K;
  for (int k = 0; k < K; k += 4) {
    v2f a, b;
    a.x = arow[k + kh];
    a.y = arow[k + kh + 1];
    b.x = B[(k + kh) * Nn + n0 + row];
    b.y = B[(k + kh + 1) * Nn + n0 + row];
    acc = __builtin_amdgcn_wmma_f32_16x16x4_f32(false, a, false, b, (short)0, acc,
                                                false, false);
  }
  int mbase = m0 + ((lane >> 4) << 3);
  int col = n0 + row;
#pragma unroll
  for (int v = 0; v < 8; ++v) C[(mbase + v) * Nn + col] = acc[v];
}

// ---------------- a_src/a_dst: wave per (node, head) ----------------
__global__ void k_asrcdst(const float* __restrict__ xp, const float* __restrict__ atts,
                          const float* __restrict__ attd, float* __restrict__ a_src,
                          float* __restrict__ a_dst) {
  int wid = (blockIdx.x * blockDim.x + threadIdx.x) >> 5;
  int lane = threadIdx.x & 31;
  if (wid >= NN * NH) return;
  int n = wid >> 2, h = wid & 3;
  const float* row = xp + n * (NH * HIDD) + h * HIDD;
  float as = 0.f, ad = 0.f;
#pragma unroll
  for (int j = 0; j < 4; ++j) {
    int c = lane + j * 32;
    float v = row[c];
    as += v * atts[h * HIDD + c];
    ad += v * attd[h * HIDD + c];
  }
  for (int off = 16; off; off >>= 1) {
    as += __shfl_xor(as, off, 32);
    ad += __shfl_xor(ad, off, 32);
  }
  if (lane == 0) { a_src[wid] = as; a_dst[wid] = ad; }
}

// ---------------- a_e = row(ea) @ v_e : wave per (edge or self-loop) ----------------
__global__ void k_ae(const float* __restrict__ ea, const float* __restrict__ selfa,
                     const float* __restrict__ ve, float* __restrict__ a_e) {
  int wid = (blockIdx.x * blockDim.x + threadIdx.x) >> 5;
  int lane = threadIdx.x & 31;
  if (wid >= E2) return;
  const float* row = (wid < EE) ? (ea + wid * HIDD) : (selfa + (wid - EE) * HIDD);
  float a0 = 0.f, a1 = 0.f, a2 = 0.f, a3 = 0.f;
#pragma unroll
  for (int j = 0; j < 4; ++j) {
    int k = lane + j * 32;
    float v = row[k];
    a0 += v * ve[k * 4 + 0];
    a1 += v * ve[k * 4 + 1];
    a2 += v * ve[k * 4 + 2];
    a3 += v * ve[k * 4 + 3];
  }
  for (int off = 16; off; off >>= 1) {
    a0 += __shfl_xor(a0, off, 32);
    a1 += __shfl_xor(a1, off, 32);
    a2 += __shfl_xor(a2, off, 32);
    a3 += __shfl_xor(a3, off, 32);
  }
  if (lane == 0) {
    a_e[wid * 4 + 0] = a0; a_e[wid * 4 + 1] = a1;
    a_e[wid * 4 + 2] = a2; a_e[wid * 4 + 3] = a3;
  }
}

// ---------------- softmax pass 1: logits + segment max ----------------
__global__ void k_logit(const int* __restrict__ ei, const float* __restrict__ a_src,
                        const float* __restrict__ a_dst, const float* __restrict__ a_e,
                        float* __restrict__ logits, unsigned* __restrict__ mbits) {
  int i = blockIdx.x * blockDim.x + threadIdx.x;
  if (i >= E2 * NH) return;
  int e = i >> 2, h = i & 3;
  int s, d;
  if (e < EE) { s = ei[e]; d = ei[EE + e]; } else { s = d = e - EE; }
  float l = leaky(a_src[s * 4 + h] + a_dst[d * 4 + h] + a_e[i], 0.2f);
  logits[i] = l;
  atomicMax(&mbits[d * 4 + h], f2o(l));
}

// ---------------- softmax pass 2: exp + segment sum ----------------
__global__ void k_expsum(const int* __restrict__ ei, float* __restrict__ logits,
                         const unsigned* __restrict__ mbits, float* __restrict__ ssum) {
  int i = blockIdx.x * blockDim.x + threadIdx.x;
  if (i >= E2 * NH) return;
  int e = i >> 2, h = i & 3;
  int d = (e < EE) ? ei[EE + e] : (e - EE);
  float ex = expf(logits[i] - o2f(mbits[d * 4 + h]));
  logits[i] = ex;
  atomicAdd(&ssum[d * 4 + h], ex);
}

// ---------------- GAT message scatter: wave per edge, head-mean fused ----------------
__global__ void k_gat_msg(const int* __restrict__ ei, const float* __restrict__ xp,
                          const float* __restrict__ ex, const float* __restrict__ ssum,
                          float* __restrict__ h) {
  int wid = (blockIdx.x * blockDim.x + threadIdx.x) >> 5;
  int lane = threadIdx.x & 31;
  if (wid >= E2) return;
  int s, d;
  if (wid < EE) { s = ei[wid]; d = ei[EE + wid]; } else { s = d = wid - EE; }
  float al0 = ex[wid * 4 + 0] / (ssum[d * 4 + 0] + 1e-16f);
  float al1 = ex[wid * 4 + 1] / (ssum[d * 4 + 1] + 1e-16f);
  float al2 = ex[wid * 4 + 2] / (ssum[d * 4 + 2] + 1e-16f);
  float al3 = ex[wid * 4 + 3] / (ssum[d * 4 + 3] + 1e-16f);
  const float* xr = xp + s * (NH * HIDD);
  float* hd = h + d * HIDD;
#pragma unroll
  for (int j = 0; j < 4; ++j) {
    int c = lane + j * 32;
    float v = 0.25f * (xr[c] * al0 + xr[HIDD + c] * al1 +
                       xr[2 * HIDD + c] * al2 + xr[3 * HIDD + c] * al3);
    atomicAdd(&hd[c], v);
  }
}

// ---------------- GCN scatter: wave per edge ----------------
__global__ void k_gcn_scatter(const int* __restrict__ ei, const float* __restrict__ hp,
                              const float* __restrict__ dinv, float* __restrict__ agg) {
  int wid = (blockIdx.x * blockDim.x + threadIdx.x) >> 5;
  int lane = threadIdx.x & 31;
  if (wid >= E2) return;
  int s, d;
  if (wid < EE) { s = ei[wid]; d = ei[EE + wid]; } else { s = d = wid - EE; }
  float ne = dinv[s] * dinv[d];
  const float* hs = hp + s * HIDD;
  float* ad = agg + d * HIDD;
#pragma unroll
  for (int j = 0; j < 4; ++j) {
    int c = lane + j * 32;
    atomicAdd(&ad[c], hs[c] * ne);
  }
}

// ---------------- GraphNorm (+leaky, +optional residual): block per graph ----------------
__global__ void k_graphnorm(const float* __restrict__ hin, const float* __restrict__ resid,
                            float* __restrict__ hout, const float* __restrict__ w,
                            const float* __restrict__ b, const float* __restrict__ ms,
                            const int* __restrict__ goff, float slope, int use_res) {
  int g = blockIdx.x, c = threadIdx.x;  // 128 threads, one channel each
  int n0 = goff[g], n1 = goff[g + 1];
  float cnt = fmaxf((float)(n1 - n0), 1.f);
  float sum = 0.f;
  for (int n = n0; n < n1; ++n) sum += hin[n * HIDD + c];
  float msm = ms[c] * (sum / cnt);
  float var = 0.f;
  for (int n = n0; n < n1; ++n) { float t = hin[n * HIDD + c] - msm; var += t * t; }
  float rs = rsqrtf(var / cnt + 1e-5f);
  float wc = w[c], bc = b[c];
  for (int n = n0; n < n1; ++n) {
    float o = leaky(wc * (hin[n * HIDD + c] - msm) * rs + bc, slope);
    if (use_res) o += resid[n * HIDD + c];
    hout[n * HIDD + c] = o;
  }
}

// ---------------- global mean pool ----------------
__global__ void k_pool(const float* __restrict__ h, const int* __restrict__ goff,
                       float* __restrict__ pooled) {
  int g = blockIdx.x, c = threadIdx.x;
  int n0 = goff[g], n1 = goff[g + 1];
  float cnt = fmaxf((float)(n1 - n0), 1.f);
  float s = 0.f;
  for (int n = n0; n < n1; ++n) s += h[n * HIDD + c];
  pooled[g * HIDD + c] = s / cnt;
}

// ---------------- MLP head: block per graph ----------------
__global__ void k_mlp(const float* __restrict__ pooled,
                      const float* __restrict__ W1, const float* __restrict__ b1,
                      const float* __restrict__ W2, const float* __restrict__ b2,
                      const float* __restrict__ W3, const float* __restrict__ b3,
                      float* __restrict__ out) {
  __shared__ float p[HIDD], z1[HIDD], z2[HIDD / 2];
  int g = blockIdx.x, t = threadIdx.x;
  p[t] = pooled[g * HIDD + t];
  __syncthreads();
  float a = b1[t];
  for (int k = 0; k < HIDD; ++k) a += p[k] * W1[k * HIDD + t];
  z1[t] = leaky(a, 0.01f);
  __syncthreads();
  if (t < HIDD / 2) {
    float a2 = b2[t];
    for (int k = 0; k < HIDD; ++k) a2 += z1[k] * W2[k * (HIDD / 2) + t];
    z2[t] = leaky(a2, 0.01f);
  }
  __syncthreads();
  if (t < 2) {
    float a3 = b3[t];
    for (int k = 0; k < HIDD / 2; ++k) a3 += z2[k] * W3[k * 2 + t];
    out[g * 2 + t] = a3;
  }
}

// ============================================================================
extern "C" void kernel_launch(void* const* d_in, const int* in_sizes, int n_in,
                              void* d_out, int out_size, void* d_ws, size_t ws_size,
                              hipStream_t stream) {
  (void)n_in; (void)out_size; (void)ws_size;
  // ---- input index map: detect flattening order by first leaf size ----
  int I_x, I_ei, I_ea, I_batch, I_gatW, I_gatWe, I_asrc, I_adst, I_aedge, I_gatb;
  int I_gcnW[4], I_gcnb[4], I_nw[5], I_nb[5], I_nms[5];
  int I_f1W, I_f1b, I_f2W, I_f2b, I_f3W, I_f3b;
  if (in_sizes[0] == NN * IND) {
    // recursive dict-insertion order
    I_x = 0; I_ei = 1; I_ea = 2; I_batch = 3;
    I_gatW = 4; I_gatWe = 5; I_asrc = 6; I_adst = 7; I_aedge = 8; I_gatb = 9;
    for (int i = 0; i < 4; ++i) { I_gcnW[i] = 10 + 2 * i; I_gcnb[i] = 11 + 2 * i; }
    for (int i = 0; i < 5; ++i) { I_nw[i] = 18 + 3 * i; I_nb[i] = 19 + 3 * i; I_nms[i] = 20 + 3 * i; }
    I_f1W = 33; I_f1b = 34; I_f2W = 35; I_f2b = 36; I_f3W = 37; I_f3b = 38;
  } else {
    // jax tree_leaves alphabetical order
    I_batch = 0; I_ea = 1; I_ei = 2;
    I_f1W = 3; I_f1b = 4; I_f2W = 5; I_f2b = 6; I_f3W = 7; I_f3b = 8;
    I_gatW = 9; I_gatWe = 10; I_adst = 11; I_aedge = 12; I_asrc = 13; I_gatb = 14;
    for (int i = 0; i < 4; ++i) { I_gcnW[i] = 15 + 2 * i; I_gcnb[i] = 16 + 2 * i; }
    for (int i = 0; i < 5; ++i) { I_nb[i] = 23 + 3 * i; I_nms[i] = 24 + 3 * i; I_nw[i] = 25 + 3 * i; }
    I_x = 38;
  }
  const float* x     = (const float*)d_in[I_x];
  const int*   ei    = (const int*)d_in[I_ei];
  const float* ea    = (const float*)d_in[I_ea];
  const int*   batch = (const int*)d_in[I_batch];
  const float* gatW  = (const float*)d_in[I_gatW];
  const float* gatWe = (const float*)d_in[I_gatWe];
  const float* atts  = (const float*)d_in[I_asrc];
  const float* attd  = (const float*)d_in[I_adst];
  const float* atte  = (const float*)d_in[I_aedge];
  const float* gatb  = (const float*)d_in[I_gatb];
  float* out = (float*)d_out;

  // ---- workspace layout ----
  char* base = (char*)d_ws;
  size_t off = 0;
  auto alloc = [&](size_t bytes) -> void* {
    void* p = base + off;
    off = (off + bytes + 255) & ~(size_t)255;
    return p;
  };
  float*    xp     = (float*)alloc((size_t)NN * NH * HIDD * 4);  // 102.4 MB
  float*    selfa  = (float*)alloc((size_t)NN * HIDD * 4);       // 25.6 MB
  float*    hbuf   = (float*)alloc((size_t)NN * HIDD * 4);       // 25.6 MB
  float*    agg    = (float*)alloc((size_t)NN * HIDD * 4);       // 25.6 MB
  float*    hp     = (float*)alloc((size_t)NN * HIDD * 4);       // 25.6 MB
  float*    a_src  = (float*)alloc((size_t)NN * NH * 4);
  float*    a_dst  = (float*)alloc((size_t)NN * NH * 4);
  float*    a_e    = (float*)alloc((size_t)E2 * NH * 4);
  float*    logits = (float*)alloc((size_t)E2 * NH * 4);
  float*    ssum   = (float*)alloc((size_t)NN * NH * 4);
  unsigned* mbits  = (unsigned*)alloc((size_t)NN * NH * 4);
  unsigned* cnt    = (unsigned*)alloc((size_t)NN * 4);
  float*    dinv   = (float*)alloc((size_t)NN * 4);
  float*    ve     = (float*)alloc((size_t)HIDD * NH * 4);
  unsigned* gcount = (unsigned*)alloc((size_t)GG * 4);
  int*      goff   = (int*)alloc((size_t)(GG + 1) * 4);
  float*    pooled = (float*)alloc((size_t)GG * HIDD * 4);

  const int T = 256;
  auto nb = [](int n, int t) { return (n + t - 1) / t; };

  // ---- init ----
  k_fill_u32<<<nb(NN, T), T, 0, stream>>>(cnt, 0u, NN);
  k_fill_u32<<<nb(GG, T), T, 0, stream>>>(gcount, 0u, GG);
  k_fill_f32<<<nb(NN * HIDD, T), T, 0, stream>>>(selfa, 0.f, NN * HIDD);
  k_fill_u32<<<nb(NN * NH, T), T, 0, stream>>>(mbits, 0u, NN * NH);
  k_fill_f32<<<nb(NN * NH, T), T, 0, stream>>>(ssum, 0.f, NN * NH);

  // ---- degrees / graph offsets ----
  k_count_dst<<<nb(EE, T), T, 0, stream>>>(ei, cnt);
  k_count_graph<<<nb(NN, T), T, 0, stream>>>(batch, gcount);
  k_scan_graph<<<1, 1, 0, stream>>>(gcount, goff);

  // ---- GAT layer ----
  k_self_accum<<<nb(EE * HIDD, T), T, 0, stream>>>(ea, ei, selfa);
  k_self_fin<<<nb(NN * HIDD, T), T, 0, stream>>>(selfa, cnt, dinv);
  k_ve<<<nb(HIDD * NH, T), T, 0, stream>>>(gatWe, atte, ve);
  {
    int waves = (NN / 16) * ((NH * HIDD) / 16);       // 3125 * 32
    k_wmma_gemm_f32<<<nb(waves * 32, T), T, 0, stream>>>(x, gatW, xp, NN, NH * HIDD, IND);
  }
  k_asrcdst<<<nb(NN * NH * 32, T), T, 0, stream>>>(xp, atts, attd, a_src, a_dst);
  k_ae<<<nb(E2 * 32, T), T, 0, stream>>>(ea, selfa, ve, a_e);
  k_logit<<<nb(E2 * NH, T), T, 0, stream>>>(ei, a_src, a_dst, a_e, logits, mbits);
  k_expsum<<<nb(E2 * NH, T), T, 0, stream>>>(ei, logits, mbits, ssum);
  k_fill_bias<<<nb(NN * HIDD, T), T, 0, stream>>>(hbuf, gatb, NN * HIDD);
  k_gat_msg<<<nb(E2 * 32, T), T, 0, stream>>>(ei, xp, logits, ssum, hbuf);
  k_graphnorm<<<GG, HIDD, 0, stream>>>(hbuf, hbuf, hbuf,
                                       (const float*)d_in[I_nw[0]], (const float*)d_in[I_nb[0]],
                                       (const float*)d_in[I_nms[0]], goff, 0.1f, 0);

  // ---- 4x GCN layers ----
  for (int l = 0; l < 4; ++l) {
    const float* Wg = (const float*)d_in[I_gcnW[l]];
    const float* bg = (const float*)d_in[I_gcnb[l]];
    int waves = (NN / 16) * (HIDD / 16);              // 3125 * 8
    k_wmma_gemm_f32<<<nb(waves * 32, T), T, 0, stream>>>(hbuf, Wg, hp, NN, HIDD, HIDD);
    k_fill_bias<<<nb(NN * HIDD, T), T, 0, stream>>>(agg, bg, NN * HIDD);
    k_gcn_scatter<<<nb(E2 * 32, T), T, 0, stream>>>(ei, hp, dinv, agg);
    k_graphnorm<<<GG, HIDD, 0, stream>>>(agg, hbuf, hbuf,
                                         (const float*)d_in[I_nw[l + 1]],
                                         (const float*)d_in[I_nb[l + 1]],
                                         (const float*)d_in[I_nms[l + 1]], goff, 0.1f, 1);
  }

  // ---- pool + MLP head ----
  k_pool<<<GG, HIDD, 0, stream>>>(hbuf, goff, pooled);
  k_mlp<<<GG, HIDD, 0, stream>>>(pooled,
                                 (const float*)d_in[I_f1W], (const float*)d_in[I_f1b],
                                 (const float*)d_in[I_f2W], (const float*)d_in[I_f2b],
                                 (const float*)d_in[I_f3W], (const float*)d_in[I_f3b], out);
}